// kNN_8796093022437
// MI455X (gfx1250) — compile-verified
//
#include <hip/hip_runtime.h>

typedef __attribute__((ext_vector_type(2))) float v2f;
typedef __attribute__((ext_vector_type(8))) float v8f;

// Problem constants from the reference
constexpr int NB    = 2;     // batches
constexpr int NPTS  = 2048;  // points per batch
constexpr int ND    = 16;    // dims
constexpr int NK    = 16;    // neighbors kept
constexpr int TILES = NPTS / 16;  // 128 column tiles

// One wave (32 threads) per workgroup: barrier ops become S_NOP (ISA §3.1),
// so the per-tile LDS ordering costs only the s_wait_dscnt the compiler
// inserts anyway, and waves never falsely synchronize with each other.
__global__ __launch_bounds__(32) void knn_wmma_kernel(const float* __restrict__ pts,
                                                      int* __restrict__ out) {
    __shared__ float s_tile[16 * 16];   // 16x16 distance tile (row-major [M][N])
    __shared__ float s_bd[32 * 17];     // 32 sorted half-row lists: distances
    __shared__ int   s_bi[32 * 17];     // 32 sorted half-row lists: indices
    __shared__ float s_qn[16];          // query row squared norms

    const int lane = threadIdx.x & 31;
    const int m    = lane & 15;       // row (or column) position within 16
    const int h    = lane >> 4;       // lane half

    const int task  = blockIdx.x;                 // 0..255
    const int b     = task / TILES;               // batch
    const int qbase = (task % TILES) * 16;        // query tile base row

    const float* batchPts = pts + (size_t)b * NPTS * ND;

    // ---- A operand (16x4 f32, chained over 4 K-chunks).
    // Lane l holds M=l%16; VGPR0 = K(2*h), VGPR1 = K(2*h+1) within each chunk.
    v2f qa[4];
    const float* qrow = batchPts + (size_t)(qbase + m) * ND;
    float qpart = 0.f;
#pragma unroll
    for (int kk = 0; kk < 4; ++kk) {
        float x = qrow[4 * kk + 2 * h + 0];
        float y = qrow[4 * kk + 2 * h + 1];
        qa[kk].x = x;
        qa[kk].y = y;
        qpart += x * x + y * y;
    }
    // Each lane half holds half the dims; combine -> full |q|^2.
    float qn_full = qpart + __shfl_xor(qpart, 16, 32);
    if (lane < 16) s_qn[m] = qn_full;

    // ---- Init all 32 per-half-row top-17 lists (sorted ascending; slot 16 = worst).
    {
        const int base = lane * 17;
        for (int j = 0; j < 17; ++j) {
            s_bd[base + j] = 3.4e38f;
            s_bi[base + j] = 0;
        }
    }
    __syncthreads();

    // Each lane caches the 8 query-row norms it needs (C/D rows M = r + 8*h).
    float qn_r[8];
#pragma unroll
    for (int r = 0; r < 8; ++r) qn_r[r] = s_qn[r + 8 * h];

    const int listBase = lane * 17;

    // ---- Stream over all 128 candidate column tiles.
    for (int ct = 0; ct < TILES; ++ct) {
        const int cbase = ct * 16;

        // B operand (4x16 f32): lane l holds column N=l%16; same K split as A.
        const float* crow = batchPts + (size_t)(cbase + m) * ND;
        v2f   ca[4];
        float cpart = 0.f;
#pragma unroll
        for (int kk = 0; kk < 4; ++kk) {
            float x = crow[4 * kk + 2 * h + 0];
            float y = crow[4 * kk + 2 * h + 1];
            ca[kk].x = x;
            ca[kk].y = y;
            cpart += x * x + y * y;
        }
        float cn = cpart + __shfl_xor(cpart, 16, 32);  // |c_n|^2 for n = lane%16

        // dot(q_m, c_n) over D=16 via 4 chained f32 WMMAs (K=4 each).
        v8f acc = {};
#pragma unroll
        for (int kk = 0; kk < 4; ++kk) {
            acc = __builtin_amdgcn_wmma_f32_16x16x4_f32(
                /*neg_a=*/false, qa[kk],
                /*neg_b=*/false, ca[kk],
                /*c_mod=*/(short)0, acc,
                /*reuse_a=*/false, /*reuse_b=*/false);
        }

        // dist^2[M][N] = |q_M|^2 + |c_N|^2 - 2*dot.  Lane holds N=m, M=r+8*h.
#pragma unroll
        for (int r = 0; r < 8; ++r) {
            float d = qn_r[r] + cn - 2.0f * acc[r];
            s_tile[(r + 8 * h) * 16 + m] = d;
        }
        __syncthreads();  // single-wave workgroup: lowers to dscnt wait + S_NOP barrier

        // All 32 lanes scan: lane (h,m) handles columns 8h..8h+7 of row m,
        // inserting into its private sorted top-17 list.
        {
            float worst = s_bd[listBase + 16];
#pragma unroll 1
            for (int nn = 0; nn < 8; ++nn) {
                const int n = 8 * h + nn;
                float d = s_tile[m * 16 + n];
                if (d < worst) {
                    int idx = cbase + n;
                    int j = 16;
                    // strict '>' keeps earlier (lower-index) entries first on
                    // ties; within a list indices are inserted in increasing order.
                    while (j > 0 && s_bd[listBase + j - 1] > d) {
                        s_bd[listBase + j] = s_bd[listBase + j - 1];
                        s_bi[listBase + j] = s_bi[listBase + j - 1];
                        --j;
                    }
                    s_bd[listBase + j] = d;
                    s_bi[listBase + j] = idx;
                    worst = s_bd[listBase + 16];
                }
            }
        }
        __syncthreads();
    }

    // ---- Merge the two half-row lists per row with exact (dist, idx)
    // lexicographic order (== top_k tie-breaking), drop slot 0 (self), emit.
    if (lane < 16) {
        const int base0 = m * 17;          // columns n%16 < 8
        const int base1 = (m + 16) * 17;   // columns n%16 >= 8
        const int row = qbase + m;
        int* o = out + (((size_t)b * NPTS + row) * NK) * 2;

        int i = 0, j = 0;
        for (int k = 0; k < 17; ++k) {   // i+j == k <= 16 at each read: in bounds
            float d0 = s_bd[base0 + i];
            float d1 = s_bd[base1 + j];
            int   x0 = s_bi[base0 + i];
            int   x1 = s_bi[base1 + j];
            bool take0 = (d0 < d1) || ((d0 == d1) && (x0 < x1));
            int sel = take0 ? x0 : x1;
            if (take0) ++i; else ++j;
            if (k > 0) {                 // k==0 is self (distance ~0)
                o[2 * (k - 1) + 0] = b;
                o[2 * (k - 1) + 1] = sel;
            }
        }
    }
}

extern "C" void kernel_launch(void* const* d_in, const int* in_sizes, int n_in,
                              void* d_out, int out_size, void* d_ws, size_t ws_size,
                              hipStream_t stream) {
    // setup_inputs order: d_in[0] = features (unused by reference), d_in[1] = points.
    const float* pts = (const float*)d_in[1];
    int* out = (int*)d_out;  // reference output dtype is int32

    const int blocks = NB * TILES;  // 256 single-wave workgroups
    knn_wmma_kernel<<<blocks, 32, 0, stream>>>(pts, out);
}